// QLayer_90778428768617
// MI455X (gfx1250) — compile-verified
//
#include <hip/hip_runtime.h>

#define NQ       14
#define DIM      (1 << NQ)
#define NLAYERS  4
#define NTHREADS 256
#define NWAVES   (NTHREADS / 32)
#define NTILES   ((DIM / 16) / 16)   // 1024 groups of 16 -> 64 tiles of 16 groups

typedef float v2f __attribute__((ext_vector_type(2)));
typedef float v8f __attribute__((ext_vector_type(8)));

// One CNOT-ring pass as a GF(2)-linear index map: state_new[k] = state_old[R(k)],
// with R(k) = F_(0,1)(F_(1,2)(...F_(13,0)(k)...)), F_(c,t): bit_t ^= bit_c.
// Qubit q lives at bit position (NQ-1-q).
__device__ __forceinline__ int r_apply(int k) {
#pragma unroll
  for (int g = NQ - 1; g >= 0; --g) {
    int c = g, t = (g + 1) % NQ;
    int pc = NQ - 1 - c, pt = NQ - 1 - t;
    k ^= ((k >> pc) & 1) << pt;
  }
  return k;
}

// XOR-combine transformed strides selected by the 4-bit element index j.
__device__ __forceinline__ int xoff(int j, int t3, int t2, int t1, int t0) {
  int r = 0;
  r ^= ((j >> 3) & 1) ? t3 : 0;
  r ^= ((j >> 2) & 1) ? t2 : 0;
  r ^= ((j >> 1) & 1) ? t1 : 0;
  r ^= ( j       & 1) ? t0 : 0;
  return r;
}

__global__ __launch_bounds__(NTHREADS, 1)
void qlayer_wmma_kernel(const float* __restrict__ x,      // (B, 14)
                        const float* __restrict__ params, // (4, 1, 14)
                        float* __restrict__ out)          // (B, 14)
{
  __shared__ float sr[DIM];                 // state real (64 KB)
  __shared__ float si[DIM];                 // state imag (64 KB)
  __shared__ float mr[256];                 // fused 16x16 gate, real part
  __shared__ float mi[256];                 // fused 16x16 gate, imag part
  __shared__ int   tlo[NLAYERS + 1][128];   // T = R^L lookup, low 7 index bits
  __shared__ int   thi[NLAYERS + 1][128];   // T = R^L lookup, high 7 index bits
  __shared__ float wcf[NLAYERS * NQ], wsf[NLAYERS * NQ];  // cos/sin(param/2)
  __shared__ float xc[NQ], xs[NQ];          // cos/sin(x/2) for this batch element
  __shared__ float red[NQ];

  const int tid  = threadIdx.x;
  const int lane = tid & 31;
  const int wave = tid >> 5;
  const int b    = blockIdx.x;

  // ---- Build permutation tables T = R^L for L = 0..NLAYERS -------------------
  for (int i = tid; i < (NLAYERS + 1) * 128; i += NTHREADS) {
    int L = i >> 7, e = i & 127;
    int a = e, h = e << 7;
    for (int l = 0; l < L; ++l) { a = r_apply(a); h = r_apply(h); }
    tlo[L][e] = a;
    thi[L][e] = h;
  }
  // ---- Rotation angles -------------------------------------------------------
  for (int i = tid; i < NLAYERS * NQ; i += NTHREADS) {
    float a = 0.5f * params[i];
    wcf[i] = cosf(a);
    wsf[i] = sinf(a);
  }
  if (tid < NQ) {
    float a = 0.5f * x[b * NQ + tid];
    xc[tid] = cosf(a);
    xs[tid] = sinf(a);
  }
  __syncthreads();

  // ---- RX angle-embedded product state: amp_k = (prod c/s) * (-i)^popcount(k)
  for (int k = tid; k < DIM; k += NTHREADS) {
    float m = 1.f;
#pragma unroll
    for (int q = 0; q < NQ; ++q)
      m *= ((k >> (NQ - 1 - q)) & 1) ? xs[q] : xc[q];
    int p = __popc(k) & 3;
    sr[k] = (p == 0) ?  m : ((p == 2) ? -m : 0.f);
    si[k] = (p == 1) ? -m : ((p == 3) ?  m : 0.f);
  }

  // ---- Layers: 4 fused 16x16 RX-group gates, CNOT ring folded into T ---------
  for (int layer = 0; layer < NLAYERS; ++layer) {
    const int* Tl = tlo[layer];
    const int* Th = thi[layer];
    for (int grp = 0; grp < 4; ++grp) {
      __syncthreads();  // previous gate's state writes + mr/mi consumers done

      int q3, q2, q1, q0;
      float c3, s3, c2, s2, c1, s1, c0, s0;
      if (grp < 3) {
        q3 = 4 * grp; q2 = 4 * grp + 1; q1 = 4 * grp + 2; q0 = 4 * grp + 3;
        c3 = wcf[layer * NQ + q3]; s3 = wsf[layer * NQ + q3];
        c2 = wcf[layer * NQ + q2]; s2 = wsf[layer * NQ + q2];
        c1 = wcf[layer * NQ + q1]; s1 = wsf[layer * NQ + q1];
        c0 = wcf[layer * NQ + q0]; s0 = wsf[layer * NQ + q0];
      } else {  // qubits 12,13 padded with identity on 0,1 (already rotated)
        q3 = 12; q2 = 13; q1 = 0; q0 = 1;
        c3 = wcf[layer * NQ + 12]; s3 = wsf[layer * NQ + 12];
        c2 = wcf[layer * NQ + 13]; s2 = wsf[layer * NQ + 13];
        c1 = 1.f; s1 = 0.f; c0 = 1.f; s0 = 0.f;
      }

      // Fused gate M[r][c] = (prod over members: equal-bit? cos : sin) * (-i)^popc(r^c)
      {
        int dx = (tid >> 4) ^ (tid & 15);
        float m = (((dx >> 3) & 1) ? s3 : c3) * (((dx >> 2) & 1) ? s2 : c2)
                * (((dx >> 1) & 1) ? s1 : c1) * ((dx & 1) ? s0 : c0);
        int p = __popc(dx) & 3;
        mr[tid] = (p == 0) ?  m : ((p == 2) ? -m : 0.f);
        mi[tid] = (p == 1) ? -m : ((p == 3) ?  m : 0.f);
      }
      __syncthreads();

      // Logical strides and their images under T (T linear over XOR)
      const int S3 = 1 << (NQ - 1 - q3), S2 = 1 << (NQ - 1 - q2);
      const int S1 = 1 << (NQ - 1 - q1), S0 = 1 << (NQ - 1 - q0);
      const int posMask = S3 | S2 | S1 | S0;
      const int tS3 = (S3 < 128) ? Tl[S3] : Th[S3 >> 7];
      const int tS2 = (S2 < 128) ? Tl[S2] : Th[S2 >> 7];
      const int tS1 = (S1 < 128) ? Tl[S1] : Th[S1 >> 7];
      const int tS0 = (S0 < 128) ? Tl[S0] : Th[S0 >> 7];

      // B operands (shared across all tiles of this gate). M symmetric -> B = M.
      // B 4x16 layout: VGPR0: lanes0-15 row k0, lanes16-31 row k0+2; VGPR1: +1 rows.
      const int bcol  = lane & 15;
      const int khalf = (lane >> 4) << 1;  // 0 or 2
      v2f bR[4], bI[4], bIn[4];
#pragma unroll
      for (int t = 0; t < 4; ++t) {
        int k0 = 4 * t + khalf;
        bR[t].x  = mr[k0 * 16 + bcol];       bR[t].y  = mr[(k0 + 1) * 16 + bcol];
        bI[t].x  = mi[k0 * 16 + bcol];       bI[t].y  = mi[(k0 + 1) * 16 + bcol];
        bIn[t].x = -bI[t].x;                 bIn[t].y = -bI[t].y;
      }
      const int toffcol = xoff(bcol, tS3, tS2, tS1, tS0);

      for (int tile = wave; tile < NTILES; tile += NWAVES) {
        // Physical base of this lane's group row (distribute 10 group-index bits
        // into the non-member bit positions, then map through T).
        int g = tile * 16 + (lane & 15);
        int base = 0, gb = g;
#pragma unroll
        for (int pb = 0; pb < NQ; ++pb) {
          if (!((posMask >> pb) & 1)) { base |= (gb & 1) << pb; gb >>= 1; }
        }
        int basep = Tl[base & 127] ^ Th[base >> 7];

        // A 16x4 layout: lane holds row M=lane&15; lanes0-15 -> K 4t,4t+1;
        // lanes16-31 -> K 4t+2,4t+3.
        v2f aR[4], aI[4];
#pragma unroll
        for (int t = 0; t < 4; ++t) {
          int k0 = 4 * t + khalf;                        // always even
          int o0 = xoff(k0, tS3, tS2, tS1, tS0);
          int o1 = o0 ^ tS0;
          aR[t].x = sr[basep ^ o0];  aR[t].y = sr[basep ^ o1];
          aI[t].x = si[basep ^ o0];  aI[t].y = si[basep ^ o1];
        }

        // out = M v : Dr = MR*vr - MI*vi ; Di = MR*vi + MI*vr  (K=16 chained)
        v8f Dr = {0.f, 0.f, 0.f, 0.f, 0.f, 0.f, 0.f, 0.f};
        v8f Di = {0.f, 0.f, 0.f, 0.f, 0.f, 0.f, 0.f, 0.f};
#pragma unroll
        for (int t = 0; t < 4; ++t) {
          Dr = __builtin_amdgcn_wmma_f32_16x16x4_f32(false, aR[t], false, bR[t],  (short)0, Dr, false, false);
          Dr = __builtin_amdgcn_wmma_f32_16x16x4_f32(false, aI[t], false, bIn[t], (short)0, Dr, false, false);
          Di = __builtin_amdgcn_wmma_f32_16x16x4_f32(false, aI[t], false, bR[t],  (short)0, Di, false, false);
          Di = __builtin_amdgcn_wmma_f32_16x16x4_f32(false, aR[t], false, bI[t],  (short)0, Di, false, false);
        }

        // D layout: VGPR v -> row M=v (lanes0-15) / M=v+8 (lanes16-31), N=lane&15
        const int mAdd = (lane >> 4) << 3;
#pragma unroll
        for (int v = 0; v < 8; ++v) {
          int bm   = __shfl(basep, v + mAdd, 32);  // base of group row (v+mAdd)
          int addr = bm ^ toffcol;
          sr[addr] = Dr[v];
          si[addr] = Di[v];
        }
      }
    }
  }

  // ---- Measurement: out[q] = sum_k (1-2*bit_q(k)) * |state[T^4(k)]|^2 --------
  __syncthreads();
  if (tid < NQ) red[tid] = 0.f;
  __syncthreads();

  const int* Tl4 = tlo[NLAYERS];
  const int* Th4 = thi[NLAYERS];
  float acc[NQ];
#pragma unroll
  for (int q = 0; q < NQ; ++q) acc[q] = 0.f;
  for (int k = tid; k < DIM; k += NTHREADS) {
    int addr = Tl4[k & 127] ^ Th4[k >> 7];
    float re = sr[addr], im = si[addr];
    float p = re * re + im * im;
#pragma unroll
    for (int q = 0; q < NQ; ++q)
      acc[q] += ((k >> (NQ - 1 - q)) & 1) ? -p : p;
  }
#pragma unroll
  for (int q = 0; q < NQ; ++q) atomicAdd(&red[q], acc[q]);
  __syncthreads();
  if (tid < NQ) out[b * NQ + tid] = red[tid];
}

extern "C" void kernel_launch(void* const* d_in, const int* in_sizes, int n_in,
                              void* d_out, int out_size, void* d_ws, size_t ws_size,
                              hipStream_t stream) {
  const float* x      = (const float*)d_in[0];   // (2048, 14) float32
  const float* params = (const float*)d_in[1];   // (4, 1, 14) float32
  float*       out    = (float*)d_out;           // (2048, 14) float32
  int batch = in_sizes[0] / NQ;                  // 2048
  qlayer_wmma_kernel<<<batch, NTHREADS, 0, stream>>>(x, params, out);
}